// GraphAttentionLayer_3736621548022
// MI455X (gfx1250) — compile-verified
//
#include <hip/hip_runtime.h>
#include <math.h>

// GAT forward, fused. B=2, T=8, N=2048, Fin=Fout=64.
// Softmax is over the T axis (size 8) per the reference quirk.
typedef __attribute__((ext_vector_type(2))) float v2f;
typedef __attribute__((ext_vector_type(8))) float v8f;

#define GAT_ALPHA 0.2f
#define GAT_NEG_INF -9.0e15f

constexpr int Bq = 2, Tq = 8, Nq = 2048, Fq = 64, BTq = Bq * Tq;

// ---------------------------------------------------------------------------
// Kernel A: Wh = h @ W (f32 WMMA 16x16x4), plus Wh1 = Wh@a1, Wh2 = Wh@a2.
// One wave per (bt, 16-row tile). A operand loaded directly in ISA layout:
//   lane<16: (M=lane, K={4kk,4kk+1}), lane>=16: (M=lane-16, K={4kk+2,4kk+3}).
// ---------------------------------------------------------------------------
__global__ __launch_bounds__(256) void gat_wh_kernel(
    const float* __restrict__ h, const float* __restrict__ W,
    const float* __restrict__ a, float* __restrict__ Wh,
    float* __restrict__ Wh1, float* __restrict__ Wh2) {
  const int wave = (blockIdx.x * blockDim.x + threadIdx.x) >> 5;
  const int lane = threadIdx.x & 31;
  const int half = lane >> 4;
  const int lm   = lane & 15;
  const int bt   = wave >> 7;          // 16 bt values, 128 m-tiles each
  const int m0   = (wave & 127) << 4;

  v8f c[4] = {};
  const float* hrow = h + (size_t)(bt * Nq + m0 + lm) * Fq;

#pragma unroll
  for (int kk = 0; kk < 16; ++kk) {
    const int k = 4 * kk + 2 * half;
    const float2 ain = *(const float2*)(hrow + k);   // 8B aligned (k even)
    v2f av; av.x = ain.x; av.y = ain.y;
#pragma unroll
    for (int nt = 0; nt < 4; ++nt) {
      v2f bv;
      bv.x = W[k * Fq + 16 * nt + lm];
      bv.y = W[(k + 1) * Fq + 16 * nt + lm];
      c[nt] = __builtin_amdgcn_wmma_f32_16x16x4_f32(
          false, av, false, bv, (short)0, c[nt], false, false);
    }
  }

  // Store Wh tile. C layout: VGPR r -> row (r + 8*half), col = 16*nt + lm.
#pragma unroll
  for (int nt = 0; nt < 4; ++nt)
#pragma unroll
    for (int r = 0; r < 8; ++r)
      Wh[(size_t)(bt * Nq + m0 + r + 8 * half) * Fq + 16 * nt + lm] = c[nt][r];

  // Per-row dots with a[:64] and a[64:], reduced across the 16 lanes of each
  // half-wave (xor masks 1..8 never cross the half boundary, bit 4).
  float a1v[4], a2v[4];
#pragma unroll
  for (int nt = 0; nt < 4; ++nt) {
    a1v[nt] = a[16 * nt + lm];
    a2v[nt] = a[64 + 16 * nt + lm];
  }
#pragma unroll
  for (int r = 0; r < 8; ++r) {
    float s1 = 0.f, s2 = 0.f;
#pragma unroll
    for (int nt = 0; nt < 4; ++nt) { s1 += c[nt][r] * a1v[nt]; s2 += c[nt][r] * a2v[nt]; }
#pragma unroll
    for (int m = 1; m <= 8; m <<= 1) {
      s1 += __shfl_xor(s1, m, 32);
      s2 += __shfl_xor(s2, m, 32);
    }
    if (lm == 0) {
      Wh1[bt * Nq + m0 + r + 8 * half] = s1;
      Wh2[bt * Nq + m0 + r + 8 * half] = s2;
    }
  }
}

// ---------------------------------------------------------------------------
// Kernel B: fused e -> leaky -> mask -> softmax(T) -> attn@Wh -> ELU.
// Workgroup = 8 waves; wave t owns time-step t for one (b, 16-row i-tile).
// e is computed per-lane already in the WMMA A-operand layout, staged to LDS
// only for the cross-t softmax statistics. attention is never materialized.
// ---------------------------------------------------------------------------
__global__ __launch_bounds__(256) void gat_attn_kernel(
    const float* __restrict__ Wh, const float* __restrict__ Wh1,
    const float* __restrict__ Wh2, const int* __restrict__ adj,
    float* __restrict__ out) {
  __shared__ float e_s[Tq][16][64];   // 32 KB
  __shared__ float m_s[16][64];       // 4 KB
  __shared__ float s_s[16][64];       // 4 KB

  const int t    = threadIdx.x >> 5;  // wave id == time step
  const int lane = threadIdx.x & 31;
  const int half = lane >> 4;
  const int lm   = lane & 15;
  const int b    = blockIdx.x >> 7;
  const int i0   = (blockIdx.x & 127) << 4;
  const int bt   = b * Tq + t;
  const int i    = i0 + lm;           // this lane's row (both halves)

  const float wh1_i = Wh1[bt * Nq + i];
  const float* wh2p = Wh2 + bt * Nq;
  const float* whp  = Wh + (size_t)bt * Nq * Fq;
  const int*   adjp = adj + (size_t)i * Nq;

  v8f acc[4] = {};
  v2f ev[16];

  for (int jt = 0; jt < Nq / 64; ++jt) {
    const int j0 = jt * 64;

    // --- e_t tile in A-operand layout: lane covers j = j0+4kk+2*half+{0,1}
#pragma unroll
    for (int kk = 0; kk < 16; ++kk) {
      const int jj = j0 + 4 * kk + 2 * half;
      const float2 w2 = *(const float2*)(wh2p + jj);       // 8B aligned
      float e0 = wh1_i + w2.x;
      float e1 = wh1_i + w2.y;
      e0 = e0 > 0.f ? e0 : GAT_ALPHA * e0;
      e1 = e1 > 0.f ? e1 : GAT_ALPHA * e1;
      const int2 ad = *(const int2*)(adjp + jj);
      if (ad.x <= 0) e0 = GAT_NEG_INF;
      if (ad.y <= 0) e1 = GAT_NEG_INF;
      ev[kk].x = e0; ev[kk].y = e1;
      e_s[t][lm][4 * kk + 2 * half]     = e0;
      e_s[t][lm][4 * kk + 2 * half + 1] = e1;
    }
    __syncthreads();

    // --- cross-t softmax stats, computed once by all 256 threads (4 (i,j)
    //     pairs each). All-masked (i,j): m = NEG_INF, s = 8 -> attn = 1/8,
    //     exactly matching jnp softmax of equal values.
    {
      const int p0 = threadIdx.x * 4;
#pragma unroll
      for (int q = 0; q < 4; ++q) {
        const int p = p0 + q, il = p >> 6, jl = p & 63;
        float m = e_s[0][il][jl];
#pragma unroll
        for (int tt = 1; tt < Tq; ++tt) m = fmaxf(m, e_s[tt][il][jl]);
        float s = 0.f;
#pragma unroll
        for (int tt = 0; tt < Tq; ++tt) s += expf(e_s[tt][il][jl] - m);
        m_s[il][jl] = m;
        s_s[il][jl] = s;
      }
    }
    __syncthreads();

    // Prefetch next Wh j-tile into the cache hierarchy.
    if (jt + 1 < Nq / 64)
      __builtin_prefetch(whp + (size_t)(j0 + 64) * Fq, 0, 3);

    // --- attn_t (registers, already A-layout) x Wh[j-tile] via f32 WMMA.
#pragma unroll
    for (int kk = 0; kk < 16; ++kk) {
      const int jl = 4 * kk + 2 * half;
      v2f av;
      av.x = expf(ev[kk].x - m_s[lm][jl])     / s_s[lm][jl];
      av.y = expf(ev[kk].y - m_s[lm][jl + 1]) / s_s[lm][jl + 1];
      const float* wrow0 = whp + (size_t)(j0 + jl) * Fq;
#pragma unroll
      for (int nt = 0; nt < 4; ++nt) {
        v2f bv;
        bv.x = wrow0[16 * nt + lm];
        bv.y = wrow0[Fq + 16 * nt + lm];
        acc[nt] = __builtin_amdgcn_wmma_f32_16x16x4_f32(
            false, av, false, bv, (short)0, acc[nt], false, false);
      }
    }
    __syncthreads();  // protect e_s/m_s/s_s before next iteration's writes
  }

  // --- ELU epilogue + store. C layout: VGPR r -> row i0+r+8*half.
#pragma unroll
  for (int nt = 0; nt < 4; ++nt)
#pragma unroll
    for (int r = 0; r < 8; ++r) {
      const float v = acc[nt][r];
      out[(size_t)(bt * Nq + i0 + r + 8 * half) * Fq + 16 * nt + lm] =
          v > 0.f ? v : expm1f(v);
    }
}

// ---------------------------------------------------------------------------
extern "C" void kernel_launch(void* const* d_in, const int* in_sizes, int n_in,
                              void* d_out, int out_size, void* d_ws, size_t ws_size,
                              hipStream_t stream) {
  const float* h   = (const float*)d_in[0];   // [2,8,2048,64]
  const float* W   = (const float*)d_in[1];   // [64,64]
  const float* a   = (const float*)d_in[2];   // [128,1]
  const int*   adj = (const int*)d_in[3];     // [2048,2048]
  float* out = (float*)d_out;                 // [2,8,2048,64]

  // Workspace: Wh (8 MB) + Wh1 (128 KB) + Wh2 (128 KB)
  float* Wh  = (float*)d_ws;
  float* Wh1 = Wh + (size_t)BTq * Nq * Fq;
  float* Wh2 = Wh1 + (size_t)BTq * Nq;

  // 2048 waves (16 bt x 128 m-tiles), 8 waves per block
  gat_wh_kernel<<<256, 256, 0, stream>>>(h, W, a, Wh, Wh1, Wh2);
  // 256 blocks (2 b x 128 i-tiles), 8 waves = 8 time steps per block
  gat_attn_kernel<<<256, 256, 0, stream>>>(Wh, Wh1, Wh2, adj, out);
}